// get_model_70403103916618
// MI455X (gfx1250) — compile-verified
//
#include <hip/hip_runtime.h>
#include <hip/hip_bf16.h>
#include <math.h>
#include <stdint.h>

// ---------------------------------------------------------------------------
// DGCNN part-seg forward for MI455X (gfx1250, wave32).
// fp32 end-to-end using V_WMMA_F32_16X16X4_F32 (matches fp32 reference data;
// ~30 GFLOP total vs ~12us of HBM traffic at 23.3 TB/s -> roughly balanced,
// so we keep fp32 and spend effort on fusing away the big tensors:
//  - c1 (B,N,1024) never materialized: max-pool fused via ordered-u32 atomicMax
//  - 192/1216-wide concats never materialized: segmented GEMM inputs
//  - GEMM staging uses GLOBAL_LOAD_ASYNC_TO_LDS_B128 (ASYNCcnt path)
// ---------------------------------------------------------------------------

typedef __attribute__((ext_vector_type(2))) float v2f;
typedef __attribute__((ext_vector_type(8))) float v8f;

#define BB 8
#define NN 2048
#define KNN 20
#define RR (BB * NN)
#define BN_EPS 1e-5f
#define SLOPE 0.2f

__device__ __forceinline__ float lrelu(float x) { return x >= 0.f ? x : SLOPE * x; }

// order-preserving float -> u32 encoding (for atomic max over floats)
__device__ __forceinline__ unsigned fenc(float x) {
  unsigned u = __float_as_uint(x);
  return (u & 0x80000000u) ? ~u : (u | 0x80000000u);
}
__device__ __forceinline__ float fdec(unsigned e) {
  return (e & 0x80000000u) ? __uint_as_float(e & 0x7FFFFFFFu) : __uint_as_float(~e);
}

__device__ __forceinline__ v8f wmma_f32(v2f a, v2f b, v8f c) {
  return __builtin_amdgcn_wmma_f32_16x16x4_f32(false, a, false, b, (short)0, c,
                                               false, false);
}

// async 16-byte global -> LDS copy (per-lane addresses), tracked by ASYNCcnt
__device__ __forceinline__ void async_copy_b128(unsigned lds_off,
                                                const float* gptr) {
  asm volatile("global_load_async_to_lds_b128 %0, %1, off" ::"v"(lds_off),
               "v"((unsigned long long)(uintptr_t)gptr)
               : "memory");
}
__device__ __forceinline__ void wait_async0() {
  asm volatile("s_wait_asynccnt 0x0" ::: "memory");
}

// ---------------------------------------------------------------------------
// (B,3,N) -> (B,N,3)
__global__ void transpose_in_kernel(const float* __restrict__ x,
                                    float* __restrict__ xt) {
  int id = blockIdx.x * blockDim.x + threadIdx.x;
  if (id >= RR * 3) return;
  int b = id / (NN * 3);
  int r = id - b * (NN * 3);
  int n = r / 3;
  int c = r - n * 3;
  xt[id] = x[((size_t)b * 3 + c) * NN + n];
}

// ---------------------------------------------------------------------------
// kNN on 3-D coords: one thread per (b,n), top-20 by neg squared distance.
__global__ void knn3_kernel(const float* __restrict__ xt, int* __restrict__ idx) {
  int row = blockIdx.x * blockDim.x + threadIdx.x;
  if (row >= RR) return;
  int b = row >> 11;
  int n = row & (NN - 1);
  const float* base = xt + (size_t)b * NN * 3;
  float cx = base[n * 3 + 0], cy = base[n * 3 + 1], cz = base[n * 3 + 2];
  float bd[KNN];
  int bi[KNN];
  for (int i = 0; i < KNN; i++) { bd[i] = -3.402823e38f; bi[i] = 0; }
  for (int m = 0; m < NN; m++) {
    float dx = base[m * 3 + 0] - cx;
    float dy = base[m * 3 + 1] - cy;
    float dz = base[m * 3 + 2] - cz;
    float nd = -(dx * dx + dy * dy + dz * dz);
    if (nd > bd[KNN - 1]) {
      int p = KNN - 1;
      while (p > 0 && bd[p - 1] < nd) { bd[p] = bd[p - 1]; bi[p] = bi[p - 1]; p--; }
      bd[p] = nd;
      bi[p] = m;
    }
  }
  for (int i = 0; i < KNN; i++) idx[(size_t)row * KNN + i] = bi[i];
}

// kNN on 64-D features: one thread per (b,n); center row cached in registers.
__global__ void knn64_kernel(const float* __restrict__ xf, int* __restrict__ idx) {
  int row = blockIdx.x * blockDim.x + threadIdx.x;
  if (row >= RR) return;
  int b = row >> 11;
  int n = row & (NN - 1);
  const float4* base = (const float4*)(xf + (size_t)b * NN * 64);
  const float4* crow = base + (size_t)n * 16;
  float4 ctr[16];
  float sqn = 0.f;
  for (int q = 0; q < 16; q++) {
    float4 v = crow[q];
    ctr[q] = v;
    sqn += v.x * v.x + v.y * v.y + v.z * v.z + v.w * v.w;
  }
  float bd[KNN];
  int bi[KNN];
  for (int i = 0; i < KNN; i++) { bd[i] = -3.402823e38f; bi[i] = 0; }
  for (int m = 0; m < NN; m++) {
    const float4* mr = base + (size_t)m * 16;
    __builtin_prefetch(base + (size_t)(m + 8) * 16, 0, 1);  // global_prefetch
    float dot = 0.f, sqm = 0.f;
    for (int q = 0; q < 16; q++) {
      float4 v = mr[q];
      dot += v.x * ctr[q].x + v.y * ctr[q].y + v.z * ctr[q].z + v.w * ctr[q].w;
      sqm += v.x * v.x + v.y * v.y + v.z * v.z + v.w * v.w;
    }
    float nd = 2.f * dot - sqn - sqm;
    if (nd > bd[KNN - 1]) {
      int p = KNN - 1;
      while (p > 0 && bd[p - 1] < nd) { bd[p] = bd[p - 1]; bi[p] = bi[p - 1]; p--; }
      bd[p] = nd;
      bi[p] = m;
    }
  }
  for (int i = 0; i < KNN; i++) idx[(size_t)row * KNN + i] = bi[i];
}

// ---------------------------------------------------------------------------
// Fused edge-conv: gather -> edge feat (LDS) -> WMMA GEMM (2C -> 64)
// -> BN+LReLU -> max over K.  One block = one batch x 4 points = 80 edges.
// Templated on C so the K-loop fully unrolls and the per-lane W column slice
// lives in registers (no guarded global loads inside the WMMA loop).
#define EC_PTS 4
#define EC_EDGES (EC_PTS * KNN)   // 80
#define EC_ES_STRIDE 132          // 128 + 4 pad (bank-conflict avoidance)

template <int C>
__global__ void __launch_bounds__(128) edgeconv_kernel(
    const float* __restrict__ xin, const int* __restrict__ idx,
    const float* __restrict__ Wm, const float* __restrict__ bias,
    const float* __restrict__ gam, const float* __restrict__ bet,
    const float* __restrict__ mean, const float* __restrict__ var,
    float* __restrict__ out) {
  constexpr int C2 = 2 * C;
  constexpr int C2p = (C2 + 7) & ~7;  // pad K-dim (e1: 6 -> 8)
  constexpr int NT = C2p / 4;         // WMMA K-steps
  __shared__ float Es[EC_EDGES * EC_ES_STRIDE];  // 80 x <=128 edge features
  __shared__ float Hs[EC_EDGES * 64];            // 80 x 64 activations
  const int b = blockIdx.y;
  const int nBase = blockIdx.x * EC_PTS;
  const int tid = threadIdx.x;

  // gather edge features into LDS: [nbr - ctr, ctr] (+ zero pad)
  for (int e = tid; e < EC_EDGES * C2p; e += 128) {
    int edge = e / C2p;
    int c = e - edge * C2p;
    int p = edge / KNN;
    int k = edge - p * KNN;
    int n = nBase + p;
    int j = idx[((size_t)b * NN + n) * KNN + k];
    float v;
    if (c < C)
      v = xin[((size_t)b * NN + j) * C + c] - xin[((size_t)b * NN + n) * C + c];
    else if (c < C2)
      v = xin[((size_t)b * NN + n) * C + (c - C)];
    else
      v = 0.f;
    Es[edge * EC_ES_STRIDE + c] = v;
  }

  // per-lane W column slice -> registers (zero-padded once, outside hot loop)
  const int lane = tid & 31;
  const int wave = tid >> 5;
  const int ln = lane & 15;
  const int hi = lane >> 4;  // 0: K pair {0,1}; 1: K pair {2,3} (16x4 A layout)
  const int col = wave * 16 + ln;
  v2f bw[NT];
#pragma unroll
  for (int t = 0; t < NT; t++) {
    int k0 = 4 * t + hi * 2;
    bw[t].x = (k0 < C2) ? Wm[(size_t)col * C2 + k0] : 0.f;
    bw[t].y = (k0 + 1 < C2) ? Wm[(size_t)col * C2 + k0 + 1] : 0.f;
  }
  const float sc = gam[col] * rsqrtf(var[col] + BN_EPS);
  const float add = (bias[col] - mean[col]) * sc + bet[col];
  __syncthreads();

  // 4 waves x (1 col tile each) x 5 row tiles of 16x16 WMMA
  for (int rt = 0; rt < 5; rt++) {
    v8f acc = {0.f, 0.f, 0.f, 0.f, 0.f, 0.f, 0.f, 0.f};
#pragma unroll
    for (int t = 0; t < NT; t++) {
      int k0 = 4 * t + hi * 2;
      v2f a;
      a.x = Es[(rt * 16 + ln) * EC_ES_STRIDE + k0];
      a.y = Es[(rt * 16 + ln) * EC_ES_STRIDE + k0 + 1];
      acc = wmma_f32(a, bw[t], acc);
    }
    // epilogue: BN + LReLU into LDS (C/D layout: VGPR j -> M=j / M=j+8)
#pragma unroll
    for (int j2 = 0; j2 < 8; j2++) {
      int rl = rt * 16 + j2 + hi * 8;
      Hs[rl * 64 + col] = lrelu(acc[j2] * sc + add);
    }
  }
  __syncthreads();

  // max over K=20 edges per point
  for (int id = tid; id < EC_PTS * 64; id += 128) {
    int p = id >> 6;
    int o = id & 63;
    float m = Hs[(p * KNN) * 64 + o];
    for (int k = 1; k < KNN; k++) m = fmaxf(m, Hs[(p * KNN + k) * 64 + o]);
    out[((size_t)b * NN + nBase + p) * 64 + o] = m;
  }
}

// ---------------------------------------------------------------------------
// Segmented GEMM + BN + LReLU.  Input row = virtual concat of up to 4 segments
// (seg0 may be per-batch broadcast -> fuses the 1216-wide concat away).
// Block: 256 threads = 8 waves; each block owns a 64x64 output tile.
// X/W tiles staged with GLOBAL_LOAD_ASYNC_TO_LDS_B128 (all chunks 16B-aligned;
// segment boundaries are multiples of 64 channels so no chunk straddles).
// mode 0: BN+LReLU store | mode 1: bias-only store | mode 2: BN+LReLU ->
//         ordered-u32 atomicMax (fused global max-pool; no 64 MB tensor).
#define MLP_LSTRIDE 68  // 272 B row stride: 16B-aligned + bank-conflict free

__global__ void __launch_bounds__(256) mlp_wmma_kernel(
    const float* __restrict__ s0, int c0, int bcast0,
    const float* __restrict__ s1, int c1,
    const float* __restrict__ s2, int c2,
    const float* __restrict__ s3, int c3,
    int Cin, const float* __restrict__ Wm, const float* __restrict__ bias,
    const float* __restrict__ gam, const float* __restrict__ bet,
    const float* __restrict__ mean, const float* __restrict__ var,
    float* __restrict__ out, unsigned* __restrict__ umax, int Cout, int mode) {
  __shared__ float Xs[64 * MLP_LSTRIDE];
  __shared__ float Ws[64 * MLP_LSTRIDE];
  const int tid = threadIdx.x;
  const int rowB = blockIdx.x * 64;
  const int colB = blockIdx.y * 64;
  const int lane = tid & 31;
  const int wave = tid >> 5;
  const int ln = lane & 15;
  const int hi = lane >> 4;
  const int o0 = c0, o1 = c0 + c1, o2 = o1 + c2;
  (void)c3;
  const int t0 = wave * 2, t1 = wave * 2 + 1;  // two 16x16 tiles per wave
  const int tr0 = (t0 >> 2) * 16, tc0 = (t0 & 3) * 16;
  const int tr1 = (t1 >> 2) * 16, tc1 = (t1 & 3) * 16;
  v8f acc0 = {0.f, 0.f, 0.f, 0.f, 0.f, 0.f, 0.f, 0.f};
  v8f acc1 = {0.f, 0.f, 0.f, 0.f, 0.f, 0.f, 0.f, 0.f};

  for (int kc = 0; kc < Cin; kc += 64) {
    // stage X tile (64 rows x 64 k) via async B128 copies, resolving segments
#pragma unroll
    for (int i = 0; i < 4; i++) {
      int chunk = i * 256 + tid;  // 0..1023, 16B each
      int r = chunk >> 4;
      int kk = (chunk & 15) * 4;
      int row = rowB + r;
      int kg = kc + kk;
      const float* gp;
      if (kg < o0)
        gp = bcast0 ? s0 + (size_t)(row >> 11) * c0 + kg
                    : s0 + (size_t)row * c0 + kg;
      else if (kg < o1)
        gp = s1 + (size_t)row * c1 + (kg - o0);
      else if (kg < o2)
        gp = s2 + (size_t)row * c2 + (kg - o1);
      else
        gp = s3 + (size_t)row * c3 + (kg - o2);
      async_copy_b128((unsigned)(uintptr_t)(&Xs[r * MLP_LSTRIDE + kk]), gp);
    }
    // stage W tile (64 cols x 64 k); clamp col past Cout (classifier: 50) —
    // pad columns hold duplicate data, mathematically separable in A*B and
    // masked at the store, so no zero-fill / no OOB reads needed.
#pragma unroll
    for (int i = 0; i < 4; i++) {
      int chunk = i * 256 + tid;
      int cl = chunk >> 4;
      int kk = (chunk & 15) * 4;
      int col = colB + cl;
      if (col >= Cout) col = Cout - 1;
      async_copy_b128((unsigned)(uintptr_t)(&Ws[cl * MLP_LSTRIDE + kk]),
                      Wm + (size_t)col * Cin + kc + kk);
    }
    wait_async0();
    __syncthreads();

#pragma unroll 4
    for (int kk = 0; kk < 64; kk += 4) {
      int k0 = kk + hi * 2;
      v2f a, bv;
      a.x = Xs[(tr0 + ln) * MLP_LSTRIDE + k0];
      a.y = Xs[(tr0 + ln) * MLP_LSTRIDE + k0 + 1];
      bv.x = Ws[(tc0 + ln) * MLP_LSTRIDE + k0];
      bv.y = Ws[(tc0 + ln) * MLP_LSTRIDE + k0 + 1];
      acc0 = wmma_f32(a, bv, acc0);
      a.x = Xs[(tr1 + ln) * MLP_LSTRIDE + k0];
      a.y = Xs[(tr1 + ln) * MLP_LSTRIDE + k0 + 1];
      bv.x = Ws[(tc1 + ln) * MLP_LSTRIDE + k0];
      bv.y = Ws[(tc1 + ln) * MLP_LSTRIDE + k0 + 1];
      acc1 = wmma_f32(a, bv, acc1);
    }
    __syncthreads();
  }

  // epilogue for both tiles
  for (int t = 0; t < 2; t++) {
    v8f acc = t ? acc1 : acc0;
    int tr = t ? tr1 : tr0;
    int tc = t ? tc1 : tc0;
    int col = colB + tc + ln;
    if (col < Cout) {
      float bv = bias[col];
      float sc = 0.f, add = 0.f;
      if (mode != 1) {
        sc = gam[col] * rsqrtf(var[col] + BN_EPS);
        add = (bv - mean[col]) * sc + bet[col];
      }
#pragma unroll
      for (int j2 = 0; j2 < 8; j2++) {
        int row = rowB + tr + j2 + hi * 8;
        float v = (mode == 1) ? (acc[j2] + bv) : lrelu(acc[j2] * sc + add);
        if (mode == 2)
          atomicMax(&umax[(size_t)(row >> 11) * Cout + col], fenc(v));
        else
          out[(size_t)row * Cout + col] = v;
      }
    }
  }
}

// ---------------------------------------------------------------------------
__global__ void maxinit_kernel(unsigned* __restrict__ umax, int n) {
  int id = blockIdx.x * blockDim.x + threadIdx.x;
  if (id < n) umax[id] = 0u;  // below every fenc() value
}
__global__ void maxdecode_kernel(const unsigned* __restrict__ umax,
                                 float* __restrict__ gmax, int n) {
  int id = blockIdx.x * blockDim.x + threadIdx.x;
  if (id < n) gmax[id] = fdec(umax[id]);
}

// log_softmax over 50 classes, write transposed (B, 50, N)
__global__ void logsoftmax_kernel(const float* __restrict__ logits,
                                  float* __restrict__ outp) {
  int row = blockIdx.x * blockDim.x + threadIdx.x;
  if (row >= RR) return;
  int b = row >> 11;
  int n = row & (NN - 1);
  const float* lr = logits + (size_t)row * 50;
  float mx = lr[0];
  for (int c = 1; c < 50; c++) mx = fmaxf(mx, lr[c]);
  float s = 0.f;
  for (int c = 0; c < 50; c++) s += __expf(lr[c] - mx);
  float ls = mx + __logf(s);
  for (int c = 0; c < 50; c++)
    outp[((size_t)b * 50 + c) * NN + n] = lr[c] - ls;
}

// (B,N,C) -> (B,C,N)
__global__ void transpose_out_kernel(const float* __restrict__ src,
                                     float* __restrict__ dst, int C) {
  int id = blockIdx.x * blockDim.x + threadIdx.x;
  if (id >= RR * C) return;
  int b = id / (NN * C);
  int r = id - b * (NN * C);
  int n = r / C;
  int c = r - n * C;
  dst[((size_t)b * C + c) * NN + n] = src[id];
}

// ---------------------------------------------------------------------------
extern "C" void kernel_launch(void* const* d_in, const int* in_sizes, int n_in,
                              void* d_out, int out_size, void* d_ws,
                              size_t ws_size, hipStream_t stream) {
  (void)in_sizes; (void)n_in; (void)out_size; (void)ws_size;
  const float* x = (const float*)d_in[0];
  // param blocks: w,b,g,be,m,v per layer
  const float* const* P = (const float* const*)d_in;
  const float* const* e1 = P + 1;
  const float* const* e2 = P + 7;
  const float* const* e3 = P + 13;
  const float* const* p1 = P + 19;
  const float* const* p2 = P + 25;
  const float* const* p3 = P + 31;
  const float* const* p4 = P + 37;
  const float* cls_w = P[43];
  const float* cls_b = P[44];

  size_t off = 0;
  auto alloc = [&](size_t bytes) -> char* {
    char* p = (char*)d_ws + off;
    off = (off + bytes + 255) & ~(size_t)255;
    return p;
  };
  float* xt = (float*)alloc((size_t)RR * 3 * 4);
  int* idx = (int*)alloc((size_t)RR * KNN * 4);
  float* x1 = (float*)alloc((size_t)RR * 64 * 4);
  float* x2 = (float*)alloc((size_t)RR * 64 * 4);
  float* x3 = (float*)alloc((size_t)RR * 64 * 4);
  unsigned* umax = (unsigned*)alloc((size_t)BB * 1024 * 4);
  float* gmax = (float*)alloc((size_t)BB * 1024 * 4);
  float* h2 = (float*)alloc((size_t)RR * 256 * 4);
  float* h3 = (float*)alloc((size_t)RR * 256 * 4);
  float* feat = (float*)alloc((size_t)RR * 128 * 4);
  float* logits = (float*)alloc((size_t)RR * 50 * 4);

  float* out = (float*)d_out;
  float* out_logp = out;                               // (B,50,N)
  float* out_x1 = out + (size_t)BB * 50 * NN;          // (B,64,N)
  float* out_x2 = out_x1 + (size_t)BB * 64 * NN;
  float* out_x3 = out_x2 + (size_t)BB * 64 * NN;
  float* out_feat = out_x3 + (size_t)BB * 64 * NN;     // (B,128,N)

  dim3 ecg(NN / EC_PTS, BB);
  dim3 mlpBlk(256);

  transpose_in_kernel<<<(RR * 3 + 255) / 256, 256, 0, stream>>>(x, xt);
  knn3_kernel<<<(RR + 127) / 128, 128, 0, stream>>>(xt, idx);
  edgeconv_kernel<3><<<ecg, 128, 0, stream>>>(xt, idx, e1[0], e1[1], e1[2],
                                              e1[3], e1[4], e1[5], x1);
  knn64_kernel<<<(RR + 127) / 128, 128, 0, stream>>>(x1, idx);
  edgeconv_kernel<64><<<ecg, 128, 0, stream>>>(x1, idx, e2[0], e2[1], e2[2],
                                               e2[3], e2[4], e2[5], x2);
  knn64_kernel<<<(RR + 127) / 128, 128, 0, stream>>>(x2, idx);
  edgeconv_kernel<64><<<ecg, 128, 0, stream>>>(x2, idx, e3[0], e3[1], e3[2],
                                               e3[3], e3[4], e3[5], x3);

  // c1: [x1|x2|x3] (192) -> 1024, BN+LReLU, fused global max pool (mode 2)
  maxinit_kernel<<<(BB * 1024 + 255) / 256, 256, 0, stream>>>(umax, BB * 1024);
  mlp_wmma_kernel<<<dim3(RR / 64, 1024 / 64), mlpBlk, 0, stream>>>(
      x1, 64, 0, x2, 64, x3, 64, x3, 0, 192, p1[0], p1[1], p1[2], p1[3], p1[4],
      p1[5], nullptr, umax, 1024, 2);
  maxdecode_kernel<<<(BB * 1024 + 255) / 256, 256, 0, stream>>>(umax, gmax,
                                                                BB * 1024);
  // c2: [gmax(bcast)|x1|x2|x3] (1216) -> 256
  mlp_wmma_kernel<<<dim3(RR / 64, 256 / 64), mlpBlk, 0, stream>>>(
      gmax, 1024, 1, x1, 64, x2, 64, x3, 64, 1216, p2[0], p2[1], p2[2], p2[3],
      p2[4], p2[5], h2, nullptr, 256, 0);
  // c3: 256 -> 256
  mlp_wmma_kernel<<<dim3(RR / 64, 256 / 64), mlpBlk, 0, stream>>>(
      h2, 256, 0, h2, 0, h2, 0, h2, 0, 256, p3[0], p3[1], p3[2], p3[3], p3[4],
      p3[5], h3, nullptr, 256, 0);
  // c4: 256 -> 128
  mlp_wmma_kernel<<<dim3(RR / 64, 128 / 64), mlpBlk, 0, stream>>>(
      h3, 256, 0, h3, 0, h3, 0, h3, 0, 256, p4[0], p4[1], p4[2], p4[3], p4[4],
      p4[5], feat, nullptr, 128, 0);
  // classifier: 128 -> 50, bias only (mode 1)
  mlp_wmma_kernel<<<dim3(RR / 64, 1), mlpBlk, 0, stream>>>(
      feat, 128, 0, feat, 0, feat, 0, feat, 0, 128, cls_w, cls_b, nullptr,
      nullptr, nullptr, nullptr, logits, nullptr, 50, 1);

  logsoftmax_kernel<<<(RR + 127) / 128, 128, 0, stream>>>(logits, out_logp);
  transpose_out_kernel<<<(RR * 64 + 255) / 256, 256, 0, stream>>>(x1, out_x1, 64);
  transpose_out_kernel<<<(RR * 64 + 255) / 256, 256, 0, stream>>>(x2, out_x2, 64);
  transpose_out_kernel<<<(RR * 64 + 255) / 256, 256, 0, stream>>>(x3, out_x3, 64);
  transpose_out_kernel<<<(RR * 128 + 255) / 256, 256, 0, stream>>>(feat, out_feat, 128);
}